// LaplacianKernel_27127013441674
// MI455X (gfx1250) — compile-verified
//
#include <hip/hip_runtime.h>
#include <math.h>

// out[i,j] = exp(-sum_k |X[i,k]-Y[j,k]| / (sigma+1e-8)), N=M=8192, d=64, fp32.
// Compute-bound (8.6G VALU f32 ops vs 256MB stores ~= 11us @ 23.3TB/s), so:
//  - 64x128 block tile, 4x8 register tile per thread (256 thr), fp32 VALU
//    sub + abs-add (2 ops/term, minimal), VOPD-pairable.
//  - LDS staging via CDNA5 GLOBAL_LOAD_ASYNC_TO_LDS_B32 (ASYNCcnt) with
//    stride-65 padding for conflict-free per-k reads.

#define D    64
#define BM   64
#define BN   128
#define TM   4
#define TN   8
#define LSTR 65      // LDS row stride in floats (bank-conflict padding)
#define NTHR 256

#if defined(__gfx1250__) && __has_builtin(__builtin_amdgcn_global_load_async_to_lds_b32)
#define USE_ASYNC_LDS 1
#endif

typedef __attribute__((address_space(1))) int as1_int;
typedef __attribute__((address_space(3))) int as3_int;

__global__ __launch_bounds__(NTHR)
void laplacian_l1_exp_kernel(const float* __restrict__ X,
                             const float* __restrict__ Y,
                             float* __restrict__ out,
                             int M, float neg_inv_sigma)
{
    __shared__ float Xs[BM * LSTR];
    __shared__ float Ys[BN * LSTR];

    const int t  = threadIdx.x;      // 0..255
    const int tx = t & 15;           // 16 col-groups of TN=8 -> 128 cols
    const int ty = t >> 4;           // 16 row-groups of TM=4 -> 64 rows

    const long long row0 = (long long)blockIdx.y * BM;
    const long long col0 = (long long)blockIdx.x * BN;

    const float* gx = X + row0 * D;  // X tile base, row-major [BM][64]
    const float* gy = Y + col0 * D;  // Y tile base, row-major [BN][64]

#if defined(USE_ASYNC_LDS)
    // Async DMA global -> LDS, b32 granularity so padded stride stays legal.
    #pragma unroll
    for (int j = 0; j < (BM * D) / NTHR; ++j) {      // 16 per thread
        int idx = j * NTHR + t;
        int r = idx >> 6, c = idx & 63;
        __builtin_amdgcn_global_load_async_to_lds_b32(
            (as1_int*)(gx + idx), (as3_int*)&Xs[r * LSTR + c], 0, 0);
    }
    #pragma unroll
    for (int j = 0; j < (BN * D) / NTHR; ++j) {      // 32 per thread
        int idx = j * NTHR + t;
        int r = idx >> 6, c = idx & 63;
        __builtin_amdgcn_global_load_async_to_lds_b32(
            (as1_int*)(gy + idx), (as3_int*)&Ys[r * LSTR + c], 0, 0);
    }
#if __has_builtin(__builtin_amdgcn_s_wait_asynccnt)
    __builtin_amdgcn_s_wait_asynccnt(0);
#else
    asm volatile("s_wait_asynccnt 0" ::: "memory");
#endif
#else
    // Fallback: synchronous staging.
    for (int j = 0; j < (BM * D) / NTHR; ++j) {
        int idx = j * NTHR + t;
        Xs[(idx >> 6) * LSTR + (idx & 63)] = gx[idx];
    }
    for (int j = 0; j < (BN * D) / NTHR; ++j) {
        int idx = j * NTHR + t;
        Ys[(idx >> 6) * LSTR + (idx & 63)] = gy[idx];
    }
#endif
    __syncthreads();

    float acc[TM][TN];
    #pragma unroll
    for (int i = 0; i < TM; ++i)
        #pragma unroll
        for (int j = 0; j < TN; ++j)
            acc[i][j] = 0.0f;

    const int xr = ty * TM;
    const int yc = tx * TN;

    #pragma unroll 8
    for (int k = 0; k < D; ++k) {
        float xv[TM], yv[TN];
        #pragma unroll
        for (int i = 0; i < TM; ++i) xv[i] = Xs[(xr + i) * LSTR + k];
        #pragma unroll
        for (int j = 0; j < TN; ++j) yv[j] = Ys[(yc + j) * LSTR + k];
        #pragma unroll
        for (int i = 0; i < TM; ++i)
            #pragma unroll
            for (int j = 0; j < TN; ++j)
                acc[i][j] += __builtin_fabsf(xv[i] - yv[j]);   // v_sub + v_add |..|
    }

    // exp(-dist/sigma) via native v_exp_f32, two b128 stores per row.
    #pragma unroll
    for (int i = 0; i < TM; ++i) {
        float* p = out + (size_t)(row0 + xr + i) * (size_t)M + (size_t)(col0 + yc);
        float4 a, b;
        a.x = __expf(acc[i][0] * neg_inv_sigma);
        a.y = __expf(acc[i][1] * neg_inv_sigma);
        a.z = __expf(acc[i][2] * neg_inv_sigma);
        a.w = __expf(acc[i][3] * neg_inv_sigma);
        b.x = __expf(acc[i][4] * neg_inv_sigma);
        b.y = __expf(acc[i][5] * neg_inv_sigma);
        b.z = __expf(acc[i][6] * neg_inv_sigma);
        b.w = __expf(acc[i][7] * neg_inv_sigma);
        ((float4*)p)[0] = a;
        ((float4*)p)[1] = b;
    }
}

extern "C" void kernel_launch(void* const* d_in, const int* in_sizes, int n_in,
                              void* d_out, int out_size, void* d_ws, size_t ws_size,
                              hipStream_t stream) {
    (void)n_in; (void)out_size; (void)d_ws; (void)ws_size;
    const float* X = (const float*)d_in[0];
    const float* Y = (const float*)d_in[1];
    float* out = (float*)d_out;

    const int N = in_sizes[0] / D;   // 8192
    const int M = in_sizes[1] / D;   // 8192
    const float neg_inv_sigma = -1.0f / (1.0f + 1e-8f);  // BANDWIDTH = 1.0

    dim3 grid(M / BN, N / BM);       // 64 x 128 = 8192 blocks
    laplacian_l1_exp_kernel<<<grid, NTHR, 0, stream>>>(X, Y, out, M, neg_inv_sigma);
}